// RelativePositionalEncoding_12670153523234
// MI455X (gfx1250) — compile-verified
//
#include <hip/hip_runtime.h>
#include <stdint.h>

// Problem constants (match reference)
#define BB 16
#define NN 2048
#define DD 1024
#define LL 4096   // MAX_LEN

typedef float v4f __attribute__((ext_vector_type(4)));

// 32-bit LDS byte offset of a shared-memory pointer, for the async-to-LDS
// instruction's VDST (LDS address) operand.
__device__ __forceinline__ uint32_t lds_off_u32(const void* p) {
    return (uint32_t)(size_t)(__attribute__((address_space(3))) const void*)p;
}

// ---------------------------------------------------------------------------
// Pass 1: per-batch min/max of timestamps -> (tmin, denom) pairs in workspace.
// 16 blocks x 256 threads; fixed 8-load per thread (trip count known to the
// compiler -> unconditional clause of loads), wave32 shuffle reduction,
// cross-wave LDS reduce.
// ---------------------------------------------------------------------------
__global__ void __launch_bounds__(256)
ts_minmax_kernel(const int* __restrict__ ts, float* __restrict__ stats) {
    __shared__ int smin[8];
    __shared__ int smax[8];

    const int b   = blockIdx.x;
    const int tid = threadIdx.x;
    const int* row = ts + b * NN;

    int vmin = 0x7fffffff;
    int vmax = (int)0x80000000;
    #pragma unroll
    for (int k = 0; k < NN / 256; ++k) {        // exactly 8 iterations
        int t = row[tid + k * 256];
        vmin = min(vmin, t);
        vmax = max(vmax, t);
    }

    // wave32 reduction
    #pragma unroll
    for (int m = 16; m >= 1; m >>= 1) {
        vmin = min(vmin, __shfl_xor(vmin, m, 32));
        vmax = max(vmax, __shfl_xor(vmax, m, 32));
    }
    const int wave = tid >> 5;
    if ((tid & 31) == 0) { smin[wave] = vmin; smax[wave] = vmax; }
    __syncthreads();

    if (tid == 0) {
        int mn = smin[0], mx = smax[0];
        #pragma unroll
        for (int w = 1; w < 8; ++w) { mn = min(mn, smin[w]); mx = max(mx, smax[w]); }
        // Match reference: f32 after conversion (ts < 2^24, exact).
        float tminf = (float)mn;
        float tmaxf = (float)mx;
        float denom = fmaxf(tmaxf - tminf, 1e-6f);
        stats[2 * b + 0] = tminf;
        stats[2 * b + 1] = denom;
    }
}

// ---------------------------------------------------------------------------
// Pass 2: out[row,:] = x[row,:] + pe[idx(row),:]
// One block per (b,n) row; 256 threads x float4 == one D=1024 row.
// The gathered pe row (idx uniform per block) is staged through LDS via the
// CDNA5 async path (global_load_async_to_lds_b128 / ASYNCcnt). Each lane
// consumes only its own 16 LDS bytes, so s_wait_asynccnt 0 is sufficient —
// no workgroup barrier needed. x/out are pure streaming traffic -> NT hints.
// ---------------------------------------------------------------------------
__global__ void __launch_bounds__(256)
gather_add_kernel(const float* __restrict__ x,
                  const float* __restrict__ pe,
                  const int*   __restrict__ ts,
                  const float* __restrict__ stats,
                  float*       __restrict__ out) {
    __shared__ v4f spe[256];

    const int row = blockIdx.x;          // 0 .. B*N-1
    const int tid = threadIdx.x;         // 0 .. 255
    const int b   = row >> 11;           // row / N

    // Index computation — exact f32 op order of the reference:
    // tn = (t - tmin) / denom * (L-1); idx = clip(trunc(tn), 0, L-1)
    const float tminf = stats[2 * b + 0];
    const float denom = stats[2 * b + 1];
    const float t  = (float)ts[row];
    const float tn = (t - tminf) / denom * (float)(LL - 1);
    int idx = (int)tn;                   // trunc toward zero (tn >= 0 here)
    idx = idx < 0 ? 0 : (idx > (LL - 1) ? (LL - 1) : idx);

    // Async copy of the gathered 4KB pe row into LDS (16B per lane, coalesced).
    const v4f* gsrc = (const v4f*)pe + (size_t)idx * (DD / 4) + tid;
    uint32_t ldst = lds_off_u32(&spe[tid]);
    asm volatile("global_load_async_to_lds_b128 %0, %1, off"
                 :: "v"(ldst), "v"((uint64_t)(size_t)gsrc)
                 : "memory");

    // Streaming load of x overlaps the async copy.
    const v4f* x4 = (const v4f*)x + (size_t)row * (DD / 4) + tid;
    v4f xv = __builtin_nontemporal_load(x4);

    // Wait for this lane's async LDS write, then consume own slot.
    asm volatile("s_wait_asynccnt 0" ::: "memory");

    v4f p = spe[tid];
    v4f o = xv + p;

    v4f* o4 = (v4f*)out + (size_t)row * (DD / 4) + tid;
    __builtin_nontemporal_store(o, o4);
}

// ---------------------------------------------------------------------------
// Host launcher
// ---------------------------------------------------------------------------
extern "C" void kernel_launch(void* const* d_in, const int* in_sizes, int n_in,
                              void* d_out, int out_size, void* d_ws, size_t ws_size,
                              hipStream_t stream) {
    const float* x     = (const float*)d_in[0];   // [B,N,D] f32
    const float* pe    = (const float*)d_in[1];   // [L,D]   f32
    const int*   ts    = (const int*)d_in[2];     // [B,N]   i32
    float*       out   = (float*)d_out;           // [B,N,D] f32
    float*       stats = (float*)d_ws;            // [B,2]  (tmin, denom)

    ts_minmax_kernel<<<BB, 256, 0, stream>>>(ts, stats);
    gather_add_kernel<<<BB * NN, 256, 0, stream>>>(x, pe, ts, stats, out);
}